// HEMoETorch_43722767073393
// MI455X (gfx1250) — compile-verified
//
#include <hip/hip_runtime.h>

#define D_DIM  1024
#define NTOK   4096          // B*S = 4*1024
#define VOCAB  50257
#define NEXP   64
#define TOPK   8
#define KC     64            // K-chunk staged in LDS

typedef float v2f __attribute__((ext_vector_type(2)));
typedef float v4f __attribute__((ext_vector_type(4)));
typedef float v8f __attribute__((ext_vector_type(8)));

__device__ __forceinline__ v8f wmma4(v2f a, v2f b, v8f c) {
  // D = A(16x4 f32) * B(4x16 f32) + C(16x16 f32)
  return __builtin_amdgcn_wmma_f32_16x16x4_f32(false, a, false, b, (short)0, c,
                                               false, false);
}

__device__ __forceinline__ v8f v8zero() {
  v8f z = {0.f, 0.f, 0.f, 0.f, 0.f, 0.f, 0.f, 0.f};
  return z;
}

// ---------------------------------------------------------------- gather + |h|^2
__global__ void k_gather(const int* __restrict__ x, const float* __restrict__ embed,
                         float* __restrict__ h, float* __restrict__ hsq) {
  int n = blockIdx.x;
  int t = threadIdx.x;                       // 256 threads, 4 floats each
  int tok = x[n];
  v4f v = *(const v4f*)(embed + (size_t)tok * D_DIM + t * 4);
  *(v4f*)(h + (size_t)n * D_DIM + t * 4) = v;
  float p = v.x * v.x + v.y * v.y + v.z * v.z + v.w * v.w;
  __shared__ float red[256];
  red[t] = p; __syncthreads();
  for (int s = 128; s > 0; s >>= 1) { if (t < s) red[t] += red[t + s]; __syncthreads(); }
  if (t == 0) hsq[n] = red[0];
}

// ---------------------------------------------------------------- |mu|^2 per expert
__global__ void k_musq(const float* __restrict__ mu, float* __restrict__ musq) {
  int e = blockIdx.x;
  int t = threadIdx.x;
  v4f v = *(const v4f*)(mu + (size_t)e * D_DIM + t * 4);
  float p = v.x * v.x + v.y * v.y + v.z * v.z + v.w * v.w;
  __shared__ float red[256];
  red[t] = p; __syncthreads();
  for (int s = 128; s > 0; s >>= 1) { if (t < s) red[t] += red[t + s]; __syncthreads(); }
  if (t == 0) musq[e] = red[0];
}

// ---------------------------------------------------------------- scores = exp kernel * charge
// One wave per 16(token) x 16(expert) tile; dot = h @ mu^T via WMMA f32.
__global__ void k_scores(const float* __restrict__ h, const float* __restrict__ mu,
                         const float* __restrict__ hsq, const float* __restrict__ musq,
                         const float* __restrict__ charge, float* __restrict__ scores) {
  int lane  = threadIdx.x;                   // 32 threads = 1 wave
  int tileM = blockIdx.x * 16;
  int tileE = blockIdx.y * 16;
  int r     = lane & 15;
  int koff  = (lane >> 4) * 2;               // K = {0,1} or {2,3}
  const float* ha = h  + (size_t)(tileM + r) * D_DIM + koff;
  const float* ba = mu + (size_t)(tileE + r) * D_DIM + koff;
  v8f acc = v8zero();
#pragma unroll 8
  for (int kb = 0; kb < D_DIM; kb += 4) {
    v2f a = *(const v2f*)(ha + kb);
    v2f b = *(const v2f*)(ba + kb);
    acc = wmma4(a, b, acc);
  }
  int mhalf = (lane >> 4) * 8;
  int e = tileE + r;
  float mq = musq[e];
  float cg = charge[e];
#pragma unroll
  for (int i = 0; i < 8; ++i) {
    int m = tileM + i + mhalf;
    float sq = hsq[m] + mq - 2.0f * acc[i];
    sq = fmaxf(sq, 0.0f);
    scores[(size_t)m * NEXP + e] = __expf(-sq * 0.125f) * cg;   // 1/(2*sigma^2)=0.125
  }
}

// ---------------------------------------------------------------- mean over tokens
__global__ void k_mean(const float* __restrict__ scores, float* __restrict__ meanv) {
  int e = blockIdx.x;
  int t = threadIdx.x;
  float s = 0.f;
  for (int n = t; n < NTOK; n += 256) s += scores[(size_t)n * NEXP + e];
  __shared__ float red[256];
  red[t] = s; __syncthreads();
  for (int k = 128; k > 0; k >>= 1) { if (t < k) red[t] += red[t + k]; __syncthreads(); }
  if (t == 0) meanv[e] = red[0] * (1.0f / (float)NTOK);
}

// ---------------------------------------------------------------- top-8 (stable, first-index on ties)
__global__ void k_topk(const float* __restrict__ meanv, int* __restrict__ topidx) {
  if (threadIdx.x != 0) return;
  float m[NEXP];
  for (int e = 0; e < NEXP; ++e) m[e] = meanv[e];
  for (int k = 0; k < TOPK; ++k) {
    int best = 0; float bv = -3.0e38f;
    for (int e = 0; e < NEXP; ++e) { float v = m[e]; if (v > bv) { bv = v; best = e; } }
    topidx[k] = best;
    m[best] = -3.0e38f;
  }
}

// ---------------------------------------------------------------- fused h_merged
// h_merged = relu(h@W1^T) + 0.3 * sum_k top_w[n,k] * (h @ Wk^T)
// Block = 256 threads (8 waves); block tile = 16 rows x 128 cols; A chunk in LDS.
__global__ void k_merged(const float* __restrict__ h,  const float* __restrict__ fw1,
                         const float* __restrict__ ew, const float* __restrict__ scores,
                         const int* __restrict__ topidx, float* __restrict__ hm) {
  __shared__ float As[16][KC + 4];           // stride 68: conflict-free 16x4 A reads
  __shared__ float tw[16][TOPK];

  int tid   = threadIdx.x;
  int lane  = tid & 31;
  int wave  = tid >> 5;
  int tileM = blockIdx.x * 16;
  int tileJ = blockIdx.y * 128 + wave * 16;

  int idx_r[TOPK];
#pragma unroll
  for (int e = 0; e < TOPK; ++e) idx_r[e] = topidx[e];
  if (tid < 128) {
    int rr = tid >> 3, e = tid & 7;
    tw[rr][e] = scores[(size_t)(tileM + rr) * NEXP + idx_r[e]];
  }

  v8f acc[1 + TOPK];
#pragma unroll
  for (int i = 0; i <= TOPK; ++i) acc[i] = v8zero();

  int r    = lane & 15;
  int koff = (lane >> 4) * 2;
  int wrow = tileJ + r;
  const float* fw1p = fw1 + (size_t)wrow * D_DIM;
  const float* ewp[TOPK];
#pragma unroll
  for (int e = 0; e < TOPK; ++e)
    ewp[e] = ew + (size_t)idx_r[e] * D_DIM * D_DIM + (size_t)wrow * D_DIM;

  for (int kb = 0; kb < D_DIM; kb += KC) {
    __syncthreads();
    {
      int lin = tid * 4; int rr = lin >> 6; int cc = lin & 63;
      v4f v = *(const v4f*)(h + (size_t)(tileM + rr) * D_DIM + kb + cc);
      *(v4f*)(&As[rr][cc]) = v;
    }
    __syncthreads();
#pragma unroll
    for (int kk = 0; kk < KC; kk += 4) {
      v2f a = *(const v2f*)(&As[r][kk + koff]);
      int dpos = kb + kk + koff;
      v2f b = *(const v2f*)(fw1p + dpos);
      acc[0] = wmma4(a, b, acc[0]);
#pragma unroll
      for (int e = 0; e < TOPK; ++e) {
        v2f be = *(const v2f*)(ewp[e] + dpos);
        acc[1 + e] = wmma4(a, be, acc[1 + e]);
      }
    }
  }

  int mhalf = (lane >> 4) * 8;
  int col   = tileJ + r;
#pragma unroll
  for (int i = 0; i < 8; ++i) {
    int mloc = i + mhalf;
    float fast = fmaxf(acc[0][i], 0.0f);
    float slow = 0.0f;
#pragma unroll
    for (int e = 0; e < TOPK; ++e) slow = fmaf(tw[mloc][e], acc[1 + e][i], slow);
    hm[(size_t)(tileM + mloc) * D_DIM + col] = fast + (1.0f - 0.7f) * slow;
  }
}

// ---------------------------------------------------------------- logits = hm @ fw2^T
// Block = 8 waves; wave tile = 16 rows x 32 cols (2 WMMA col-fragments reusing A);
// block tile = 16 x 256. Output streamed with non-temporal stores so the 824 MB
// logits write does not evict L2-resident fw2 (206 MB).
__global__ void k_logits(const float* __restrict__ hm, const float* __restrict__ fw2,
                         float* __restrict__ out) {
  __shared__ float As[16][KC + 4];
  int tid   = threadIdx.x;
  int lane  = tid & 31;
  int wave  = tid >> 5;
  int tileM = blockIdx.x * 16;
  int tileJ = blockIdx.y * 256 + wave * 32;
  int r     = lane & 15;
  int koff  = (lane >> 4) * 2;
  int col0  = tileJ + r;
  int col1  = tileJ + 16 + r;
  int wrow0 = col0 < VOCAB ? col0 : VOCAB - 1;    // clamp: no OOB weight reads
  int wrow1 = col1 < VOCAB ? col1 : VOCAB - 1;
  const float* bp0 = fw2 + (size_t)wrow0 * D_DIM;
  const float* bp1 = fw2 + (size_t)wrow1 * D_DIM;

  v8f acc0 = v8zero();
  v8f acc1 = v8zero();
  for (int kb = 0; kb < D_DIM; kb += KC) {
    __syncthreads();
    {
      int lin = tid * 4; int rr = lin >> 6; int cc = lin & 63;
      v4f v = *(const v4f*)(hm + (size_t)(tileM + rr) * D_DIM + kb + cc);
      *(v4f*)(&As[rr][cc]) = v;
    }
    __syncthreads();
    if (kb + KC < D_DIM) {                        // pull next B chunk ahead of use
      __builtin_prefetch(bp0 + kb + KC + koff, 0, 3);
      __builtin_prefetch(bp1 + kb + KC + koff, 0, 3);
    }
#pragma unroll
    for (int kk = 0; kk < KC; kk += 4) {
      v2f a  = *(const v2f*)(&As[r][kk + koff]);
      int dpos = kb + kk + koff;
      v2f b0 = *(const v2f*)(bp0 + dpos);
      v2f b1 = *(const v2f*)(bp1 + dpos);
      acc0 = wmma4(a, b0, acc0);
      acc1 = wmma4(a, b1, acc1);
    }
  }

  int mhalf = (lane >> 4) * 8;
  if (col0 < VOCAB) {
#pragma unroll
    for (int i = 0; i < 8; ++i)
      __builtin_nontemporal_store(acc0[i], &out[(size_t)(tileM + i + mhalf) * VOCAB + col0]);
  }
  if (col1 < VOCAB) {
#pragma unroll
    for (int i = 0; i < 8; ++i)
      __builtin_nontemporal_store(acc1[i], &out[(size_t)(tileM + i + mhalf) * VOCAB + col1]);
  }
}

// ---------------------------------------------------------------- launch
extern "C" void kernel_launch(void* const* d_in, const int* in_sizes, int n_in,
                              void* d_out, int out_size, void* d_ws, size_t ws_size,
                              hipStream_t stream) {
  const int*   x      = (const int*)  d_in[0];
  const float* embed  = (const float*)d_in[1];
  const float* fw1    = (const float*)d_in[2];
  const float* fw2    = (const float*)d_in[3];
  const float* mu     = (const float*)d_in[4];
  const float* ew     = (const float*)d_in[5];
  const float* charge = (const float*)d_in[6];
  float* out = (float*)d_out;

  // workspace layout (floats): h | hm | scores | hsq | musq | mean | topidx
  float* h      = (float*)d_ws;
  float* hm     = h      + (size_t)NTOK * D_DIM;
  float* scores = hm     + (size_t)NTOK * D_DIM;
  float* hsq    = scores + (size_t)NTOK * NEXP;
  float* musq   = hsq    + NTOK;
  float* meanv  = musq   + NEXP;
  int*   topidx = (int*)(meanv + NEXP);

  k_gather<<<NTOK, 256, 0, stream>>>(x, embed, h, hsq);
  k_musq  <<<NEXP, 256, 0, stream>>>(mu, musq);

  dim3 gs(NTOK / 16, NEXP / 16);
  k_scores<<<gs, 32, 0, stream>>>(h, mu, hsq, musq, charge, scores);

  k_mean<<<NEXP, 256, 0, stream>>>(scores, meanv);
  k_topk<<<1, 32, 0, stream>>>(meanv, topidx);

  dim3 gm(NTOK / 16, D_DIM / 128);
  k_merged<<<gm, 256, 0, stream>>>(h, fw1, ew, scores, topidx, hm);

  dim3 gl(NTOK / 16, (VOCAB + 255) / 256);
  k_logits<<<gl, 256, 0, stream>>>(hm, fw2, out);
}